// EstimationDelta_63909113364750
// MI455X (gfx1250) — compile-verified
//
#include <hip/hip_runtime.h>

typedef __attribute__((ext_vector_type(2))) float v2f;
typedef __attribute__((ext_vector_type(4))) float v4f;
typedef __attribute__((ext_vector_type(8))) float v8f;

static constexpr int WIMG   = 1024;        // image width
static constexpr int HWPIX  = 1024 * 1024; // H*W per (b,c)
static constexpr int NFRAME = 16;

// Normalized 1D Gaussian, ksize=5, sigma=3 (matches cv2.getGaussianKernel(5,3))
// w = {0.17820365, 0.21052364, 0.22254543, 0.21052364, 0.17820365}
__device__ __forceinline__ float gw5(int d) {
    float w = 0.f;
    w = (d == 0 || d == 4) ? 0.17820365f : w;
    w = (d == 1 || d == 3) ? 0.21052364f : w;
    w = (d == 2)           ? 0.22254543f : w;
    return w;
}

// ---------------------------------------------------------------------------
// Kernel 0: init global min/max (uint bit-pattern order valid for floats >= 0)
// ---------------------------------------------------------------------------
__global__ void k_init(unsigned* mm) {
    mm[0] = 0x7F800000u; // +inf  -> min accumulator
    mm[1] = 0x00000000u; // 0.0   -> max accumulator
}

// ---------------------------------------------------------------------------
// Kernel 1: temporal stats -> delta.  Streams 256 MB of x once (NT loads).
// Each thread owns 4 consecutive pixels (float4 / b128 loads per frame).
// ---------------------------------------------------------------------------
__global__ __launch_bounds__(256) void k_delta(const float* __restrict__ x,
                                               float* __restrict__ delta,
                                               int npix) {
    int t = blockIdx.x * blockDim.x + threadIdx.x;
    int p = t << 2;
    if (p >= npix) return;
    int b   = p >> 20;            // / HWPIX
    int rem = p & (HWPIX - 1);

    const v4f* x4 = (const v4f*)x;
    v4f sum  = {0.f, 0.f, 0.f, 0.f};
    v4f tot  = {0.f, 0.f, 0.f, 0.f};
    v4f prev = {0.f, 0.f, 0.f, 0.f};
#pragma unroll
    for (int f = 0; f < NFRAME; ++f) {
        int base = (((b << 4) + f) << 20) + rem;      // float index
        v4f v = __builtin_nontemporal_load(&x4[base >> 2]);
        v4f o;
        o.x = floorf(v.x * 255.f); o.y = floorf(v.y * 255.f);
        o.z = floorf(v.z * 255.f); o.w = floorf(v.w * 255.f);
        sum += o;
        if (f) {
            v4f d = o - prev;
            d.x = fabsf(d.x); d.y = fabsf(d.y);
            d.z = fabsf(d.z); d.w = fabsf(d.w);
            tot += d;
        }
        prev = o;
    }
    v4f mean = sum * (1.f / (float)NFRAME);
    v4f dl   = (tot * 1000.f) / (mean * mean);
    *(v4f*)(delta + p) = dl;
}

// ---------------------------------------------------------------------------
// Kernel 2: 5x5 separable Gaussian blur of the [rows=4096, 1024] field via
// V_WMMA_F32_16X16X4_F32 (banded-matrix formulation), reflect-101 borders,
// plus wave-level min/max reduced into global atomics.
// One wave32 per 16x16 output tile; EXEC is all-ones around every WMMA.
// The banded weight operands are hoisted: chunk c's weight pair is identical
// for the horizontal-pass B matrix and the vertical-pass A matrix.
// ---------------------------------------------------------------------------
__global__ __launch_bounds__(32) void k_blur(const float* __restrict__ delta,
                                             float* __restrict__ draw,
                                             unsigned* __restrict__ mm,
                                             int rows) {
    __shared__ float sIn[20 * 20];   // input halo tile
    __shared__ float sHz[20 * 16];   // horizontally-blurred intermediate

    const int lane = threadIdx.x;    // 0..31
    const int half = lane >> 4;      // 0 | 1  (lane group)
    const int mn   = lane & 15;      // row (M) or col (N) within tile
    const int r0   = blockIdx.y << 4;
    const int c0   = blockIdx.x << 4;

    // Load 20x20 window with reflect-101 at the global [rows, WIMG] edges.
    for (int i = lane; i < 400; i += 32) {
        int rr = i / 20, cc = i - rr * 20;
        int gr = r0 - 2 + rr;
        int gc = c0 - 2 + cc;
        gr = gr < 0 ? -gr : (gr >= rows ? 2 * rows - 2 - gr : gr);
        gc = gc < 0 ? -gc : (gc >= WIMG ? 2 * WIMG - 2 - gc : gc);
        sIn[i] = delta[gr * WIMG + gc];
    }

    // Precompute banded Gaussian operands once per wave: for chunk c,
    // element k of the K=4 slab maps to weight w[i - mn], i = 4c + 2*half + k.
    // Used as B (horizontal: Wh[i,n]=w[i-n]) and as A (vertical: Wv[m,i]=w[i-m]).
    v2f wk[5];
#pragma unroll
    for (int c = 0; c < 5; ++c) {
        int k0 = 4 * c + 2 * half;
        wk[c].x = gw5(k0 - mn);
        wk[c].y = gw5(k0 + 1 - mn);
    }
    __syncthreads();

    // Horizontal pass: Hz(20x16) = In(20x20) x Wh(20x16), done as two
    // overlapping 16-row groups of five K=4 WMMA accumulations each.
    // A layout (16x4 f32): VGPR0 <-> K = 2*half, VGPR1 <-> K = 2*half+1.
#pragma unroll
    for (int g = 0; g <= 4; g += 4) {
        v8f acc = {0.f, 0.f, 0.f, 0.f, 0.f, 0.f, 0.f, 0.f};
#pragma unroll
        for (int c = 0; c < 5; ++c) {
            int k0 = 4 * c + 2 * half;
            v2f a;
            a.x = sIn[(g + mn) * 20 + k0];
            a.y = sIn[(g + mn) * 20 + k0 + 1];
            acc = __builtin_amdgcn_wmma_f32_16x16x4_f32(
                false, a, false, wk[c], (short)0, acc, false, false);
        }
        // D layout: lanes<16 -> M=r, lanes>=16 -> M=8+r
#pragma unroll
        for (int r = 0; r < 8; ++r)
            sHz[(g + half * 8 + r) * 16 + mn] = acc[r];
    }
    __syncthreads();

    // Vertical pass: Out(16x16) = Wv(16x20) x Hz(20x16), five K=4 WMMAs.
    v8f o = {0.f, 0.f, 0.f, 0.f, 0.f, 0.f, 0.f, 0.f};
#pragma unroll
    for (int c = 0; c < 5; ++c) {
        int k0 = 4 * c + 2 * half;
        v2f b;
        b.x = sHz[(k0    ) * 16 + mn];
        b.y = sHz[(k0 + 1) * 16 + mn];
        o = __builtin_amdgcn_wmma_f32_16x16x4_f32(
            false, wk[c], false, b, (short)0, o, false, false);
    }

    // Write tile + track min/max (values are >= 0).
    float tmin = __builtin_inff();
    float tmax = 0.f;
#pragma unroll
    for (int r = 0; r < 8; ++r) {
        int row = r0 + half * 8 + r;
        float v = o[r];
        draw[row * WIMG + c0 + mn] = v;
        tmin = fminf(tmin, v);
        tmax = fmaxf(tmax, v);
    }
#pragma unroll
    for (int off = 16; off > 0; off >>= 1) {
        tmin = fminf(tmin, __shfl_xor(tmin, off, 32));
        tmax = fmaxf(tmax, __shfl_xor(tmax, off, 32));
    }
    if (lane == 0) {
        atomicMin(mm,     __float_as_uint(tmin));
        atomicMax(mm + 1, __float_as_uint(tmax));
    }
}

// ---------------------------------------------------------------------------
// Kernel 3: normalize + threshold + emit the 5 outputs (80 MB, NT stores).
// ---------------------------------------------------------------------------
__global__ __launch_bounds__(256) void k_final(const float* __restrict__ rf,
                                               const float* __restrict__ mf,
                                               const float* __restrict__ thr,
                                               const float* __restrict__ draw,
                                               const unsigned* __restrict__ mm,
                                               float* __restrict__ out,
                                               int npix) {
    int t = blockIdx.x * blockDim.x + threadIdx.x;
    int p = t << 2;
    if (p >= npix) return;

    float dmin = __uint_as_float(mm[0]);
    float dmax = __uint_as_float(mm[1]);
    float sc   = 255.f / (dmax - dmin);
    float mt   = *thr;

    v4f r = *(const v4f*)(rf + p);
    v4f m = *(const v4f*)(mf + p);
    v4f d = *(const v4f*)(draw + p);

    v4f rfi, mfi, cout, dout, tout;
#pragma unroll
    for (int i = 0; i < 4; ++i) {
        float rv = floorf(r[i] * 255.f);
        float mv = floorf(m[i] * 255.f);
        float dv = (d[i] - dmin) * sc;
        bool  mk = dv >= mt;
        rfi[i]  = rv;
        mfi[i]  = mv;
        dout[i] = dv;
        cout[i] = mk ? mv : rv;
        tout[i] = mk ? 255.f : 0.f;
    }
    __builtin_nontemporal_store(mfi,  (v4f*)(out + 0 * (size_t)npix + p));
    __builtin_nontemporal_store(rfi,  (v4f*)(out + 1 * (size_t)npix + p));
    __builtin_nontemporal_store(cout, (v4f*)(out + 2 * (size_t)npix + p));
    __builtin_nontemporal_store(dout, (v4f*)(out + 3 * (size_t)npix + p));
    __builtin_nontemporal_store(tout, (v4f*)(out + 4 * (size_t)npix + p));
}

// ---------------------------------------------------------------------------
extern "C" void kernel_launch(void* const* d_in, const int* in_sizes, int n_in,
                              void* d_out, int out_size, void* d_ws, size_t ws_size,
                              hipStream_t stream) {
    const float* x   = (const float*)d_in[0];  // [4,1,16,1024,1024] f32
    const float* rf  = (const float*)d_in[1];  // [4,1,1024,1024]    f32
    const float* mf  = (const float*)d_in[2];  // [4,1,1024,1024]    f32
    const float* thr = (const float*)d_in[3];  // scalar f32 (50.0)
    // d_in[4] = n_frames (16), baked into the kernels

    const int npix = in_sizes[1];              // 4*1*1024*1024
    const int rows = npix / WIMG;              // 4096

    float*    delta = (float*)d_ws;            // npix f32
    float*    draw  = delta + npix;            // npix f32
    unsigned* mm    = (unsigned*)(draw + npix);// 2 x u32

    k_init<<<1, 1, 0, stream>>>(mm);

    int vthreads = npix / 4;
    k_delta<<<dim3((vthreads + 255) / 256), 256, 0, stream>>>(x, delta, npix);

    dim3 bgrid(WIMG / 16, rows / 16);          // 64 x 256 tiles
    k_blur<<<bgrid, 32, 0, stream>>>(delta, draw, mm, rows);

    k_final<<<dim3((vthreads + 255) / 256), 256, 0, stream>>>(
        rf, mf, thr, draw, mm, (float*)d_out, npix);
}